// Model_12206297055267
// MI455X (gfx1250) — compile-verified
//
#include <hip/hip_runtime.h>
#include <math.h>

// ============================================================================
// CDNA5 (gfx1250) implementation of the p4-CNN + capsule routing model.
// fp32 GEMM path uses V_WMMA_F32_16X16X4_F32 (wave32, 16x16 tile, K step 4),
// one wave computes a 16(M) x 64(N) output tile (4 accumulators, A reused 4x).
// Activations stored group-minor: channel index = c*4 + r  (c = "Co" channel,
// r = rotation). With expanded weights this makes every p4conv a plain conv
// whose output layout equals the next layer's input layout.
// Expanded weights are stored K-major/transposed: wexpT[kk][OC] for coalesced
// B-operand loads. Conv kernel is templated on filter size K (kills div/mod
// in the hot loop) and on PAD (kills A-load bounds checks for pad==0 convs,
// which are the two dominant capsule GEMMs).
// ============================================================================

typedef __attribute__((ext_vector_type(2))) float v2f;
typedef __attribute__((ext_vector_type(8))) float v8f;

__device__ __forceinline__ float selu_f(float x){
  const float a = 1.6732632423543772f, s = 1.0507009873554805f;
  return x > 0.f ? s * x : s * a * (expf(x) - 1.f);
}

// ---------------- weight expansion (writes transposed [kk][OC]) ----------------
// dest row kk = ky*K+kx, col o = co*4+ro ; value = rot90^ro(w[co])[ky][kx]
__global__ void k_expand_z2(const float* __restrict__ w, float* __restrict__ wexpT,
                            int Co, int K){
  int idx = blockIdx.x * blockDim.x + threadIdx.x;
  int total = Co * 4 * K * K;
  if (idx >= total) return;
  int kx = idx % K; int t = idx / K;
  int ky = t % K;   t /= K;
  int ro = t & 3;   int co = t >> 2;
  int sy, sx;
  switch (ro){
    case 0:  sy = ky;         sx = kx;         break;
    case 1:  sy = kx;         sx = K - 1 - ky; break;
    case 2:  sy = K - 1 - ky;  sx = K - 1 - kx; break;
    default: sy = K - 1 - kx;  sx = ky;         break;
  }
  int o = co * 4 + ro;
  int kk = ky * K + kx;
  wexpT[(long)kk * (Co * 4) + o] = w[(co * K + sy) * K + sx];
}

// dest row kk = ((ci*4+s)*K+ky)*K+kx, col o = co*4+ro
// value = rot90^ro( w[co][ci][(s-ro)&3] )[ky][kx]
__global__ void k_expand_p4(const float* __restrict__ w, float* __restrict__ wexpT,
                            int Co, int Ci, int K){
  int idx = blockIdx.x * blockDim.x + threadIdx.x;
  int total = Co * 4 * Ci * 4 * K * K;
  if (idx >= total) return;
  int kx = idx % K; int t = idx / K;
  int ky = t % K;   t /= K;
  int s  = t & 3;   t >>= 2;
  int ci = t % Ci;  t /= Ci;
  int ro = t & 3;   int co = t >> 2;
  int sy, sx;
  switch (ro){
    case 0:  sy = ky;         sx = kx;         break;
    case 1:  sy = kx;         sx = K - 1 - ky; break;
    case 2:  sy = K - 1 - ky;  sx = K - 1 - kx; break;
    default: sy = K - 1 - kx;  sx = ky;         break;
  }
  int ss = (s - ro) & 3;
  int o = co * 4 + ro;
  int cin = ci * 4 + s;
  int kk = (cin * K + ky) * K + kx;
  wexpT[(long)kk * (Co * 4) + o] = w[(((co * Ci + ci) * 4 + ss) * K + sy) * K + sx];
}

// ---------------- WMMA implicit-GEMM conv ----------------
// in  : [BT][CIN][H][W]   wexpT: [CIN*K*K][OC]   out: [BT][OC][Ho][Wo]
// grid = (M/16, OC/64), block = 32 (one wave -> 16x64 D tile, 4 accumulators).
// A (16x4 f32): lanes 0-15 -> M, vgpr0=K0 vgpr1=K1 ; lanes 16-31 -> K2,K3
// B (4x16 f32): lanes 0-15 -> N, vgpr0=K0 vgpr1=K1 ; lanes 16-31 -> K2,K3
// D (16x16 f32): vgpr g: lanes0-15 M=g, lanes16-31 M=g+8 ; N = lane&15
template<int K, int PAD>
__global__ __launch_bounds__(32)
void k_conv_wmma(const float* __restrict__ in, const float* __restrict__ wexpT,
                 const float* __restrict__ bias, float* __restrict__ out,
                 int BT, int CIN, int H, int W, int OC,
                 int stride, int pad, int Ho, int Wo, int fuse_selu){
  const int t = threadIdx.x;                 // 0..31
  const int Mtot = BT * Ho * Wo;
  const int m = blockIdx.x * 16 + (t & 15);
  int bt, ho, wo;
  {
    int mm = (m < Mtot) ? m : 0;
    wo = mm % Wo; int q = mm / Wo; ho = q % Ho; bt = q / Ho;
  }
  const int hbase = ho * stride - pad;
  const int wbase = wo * stride - pad;
  const float* __restrict__ inb = in + (long)bt * CIN * H * W;
  const int n0 = blockIdx.y * 64 + (t & 15);
  const int Ktot = CIN * K * K;
  const int kpair = (t >> 4) << 1;           // 0 or 2
  const v8f z8 = {0.f, 0.f, 0.f, 0.f, 0.f, 0.f, 0.f, 0.f};
  v8f acc[4] = {z8, z8, z8, z8};
  const int Kmain = Ktot & ~3;

  int kb = 0;
  for (; kb < Kmain; kb += 4){
    v2f a, b0, b1, b2, b3;
#pragma unroll
    for (int j = 0; j < 2; ++j){
      const int kk = kb + kpair + j;
      const int kx = kk % K;                 // compile-time K -> mul/shift
      const int q  = kk / K;
      const int ky = q % K;
      const int cin = q / K;
      const int hi = hbase + ky;
      const int wi = wbase + kx;
      float av;
      if (PAD){
        av = 0.f;
        if (hi >= 0 && hi < H && wi >= 0 && wi < W)
          av = inb[((long)cin * H + hi) * W + wi];
      } else {
        av = inb[((long)cin * H + hi) * W + wi];   // pad==0: always in bounds
      }
      a[j] = av;
      const float* __restrict__ wr = wexpT + (long)kk * OC + n0;
      b0[j] = wr[0]; b1[j] = wr[16]; b2[j] = wr[32]; b3[j] = wr[48];
    }
    acc[0] = __builtin_amdgcn_wmma_f32_16x16x4_f32(false, a, false, b0, (short)0, acc[0], false, false);
    acc[1] = __builtin_amdgcn_wmma_f32_16x16x4_f32(false, a, false, b1, (short)0, acc[1], false, false);
    acc[2] = __builtin_amdgcn_wmma_f32_16x16x4_f32(false, a, false, b2, (short)0, acc[2], false, false);
    acc[3] = __builtin_amdgcn_wmma_f32_16x16x4_f32(false, a, false, b3, (short)0, acc[3], false, false);
  }
  if (kb < Ktot){                            // ragged tail (conv1: Ktot = 9)
    const v2f z2 = {0.f, 0.f};
    v2f a = z2, b0 = z2, b1 = z2, b2 = z2, b3 = z2;
#pragma unroll
    for (int j = 0; j < 2; ++j){
      const int kk = kb + kpair + j;
      if (kk < Ktot){
        const int kx = kk % K;
        const int q  = kk / K;
        const int ky = q % K;
        const int cin = q / K;
        const int hi = hbase + ky;
        const int wi = wbase + kx;
        float av = 0.f;
        if (!PAD || (hi >= 0 && hi < H && wi >= 0 && wi < W))
          av = inb[((long)cin * H + hi) * W + wi];
        a[j] = av;
        const float* __restrict__ wr = wexpT + (long)kk * OC + n0;
        b0[j] = wr[0]; b1[j] = wr[16]; b2[j] = wr[32]; b3[j] = wr[48];
      }
    }
    acc[0] = __builtin_amdgcn_wmma_f32_16x16x4_f32(false, a, false, b0, (short)0, acc[0], false, false);
    acc[1] = __builtin_amdgcn_wmma_f32_16x16x4_f32(false, a, false, b1, (short)0, acc[1], false, false);
    acc[2] = __builtin_amdgcn_wmma_f32_16x16x4_f32(false, a, false, b2, (short)0, acc[2], false, false);
    acc[3] = __builtin_amdgcn_wmma_f32_16x16x4_f32(false, a, false, b3, (short)0, acc[3], false, false);
  }

  // store 4 tiles of D
  float bval[4];
#pragma unroll
  for (int tn = 0; tn < 4; ++tn){
    int col = blockIdx.y * 64 + tn * 16 + (t & 15);
    bval[tn] = bias ? bias[col >> 2] : 0.f;
  }
  const int rbase = (t >> 4) * 8;
#pragma unroll
  for (int g = 0; g < 8; ++g){
    int row = blockIdx.x * 16 + rbase + g;
    if (row >= Mtot) continue;
    int wo2 = row % Wo; int q = row / Wo; int ho2 = q % Ho; int bt2 = q / Ho;
    long obase = (((long)bt2 * OC) * Ho + ho2) * Wo + wo2;
#pragma unroll
    for (int tn = 0; tn < 4; ++tn){
      int col = blockIdx.y * 64 + tn * 16 + (t & 15);
      float v = acc[tn][g] + bval[tn];
      if (fuse_selu) v = selu_f(v);
      out[obase + (long)col * Ho * Wo] = v;
    }
  }
}

// ---------------- fused BN3d (per-"co" channel over BT,4,H,W) ----------------
// x: [BT][Cco*4][HW]; block = co; channels co*4..co*4+3 are contiguous.
__global__ void k_bn_fused(float* __restrict__ x, const float* __restrict__ g,
                           const float* __restrict__ b, int BT, int Cco, int HW,
                           int fuse_selu){
  const int co = blockIdx.x;
  const int C4 = Cco * 4;
  const int FW = 4 * HW;
  const long n = (long)BT * FW;
  float s0 = 0.f, s1 = 0.f;
  for (long i = threadIdx.x; i < n; i += blockDim.x){
    long bt = i / FW; long rem = i - bt * FW;
    float v = x[(bt * C4 + (long)co * 4) * HW + rem];
    s0 += v; s1 += v * v;
  }
  __shared__ float sh0[256], sh1[256];
  sh0[threadIdx.x] = s0; sh1[threadIdx.x] = s1; __syncthreads();
  for (int o = blockDim.x >> 1; o > 0; o >>= 1){
    if ((int)threadIdx.x < o){
      sh0[threadIdx.x] += sh0[threadIdx.x + o];
      sh1[threadIdx.x] += sh1[threadIdx.x + o];
    }
    __syncthreads();
  }
  const float mean = sh0[0] / (float)n;
  const float var  = sh1[0] / (float)n - mean * mean;
  const float rstd = rsqrtf(var + 1e-5f);
  const float gg = g[co] * rstd;
  const float bb = b[co] - gg * mean;
  for (long i = threadIdx.x; i < n; i += blockDim.x){
    long bt = i / FW; long rem = i - bt * FW;
    long off = (bt * C4 + (long)co * 4) * HW + rem;
    float y = gg * x[off] + bb;
    if (fuse_selu) y = selu_f(y);
    x[off] = y;
  }
}

// per-batch layernorm with per-position affine (prim layer)
__global__ void k_layernorm(float* __restrict__ x, const float* __restrict__ g,
                            const float* __restrict__ b, long N){
  const long base = (long)blockIdx.x * N;
  float s0 = 0.f, s1 = 0.f;
  for (long i = threadIdx.x; i < N; i += blockDim.x){
    float v = x[base + i]; s0 += v; s1 += v * v;
  }
  __shared__ float sh0[256], sh1[256];
  sh0[threadIdx.x] = s0; sh1[threadIdx.x] = s1; __syncthreads();
  for (int o = blockDim.x >> 1; o > 0; o >>= 1){
    if ((int)threadIdx.x < o){
      sh0[threadIdx.x] += sh0[threadIdx.x + o];
      sh1[threadIdx.x] += sh1[threadIdx.x + o];
    }
    __syncthreads();
  }
  const float mean = sh0[0] / (float)N;
  const float var  = sh1[0] / (float)N - mean * mean;
  const float rstd = rsqrtf(var + 1e-5f);
  for (long i = threadIdx.x; i < N; i += blockDim.x)
    x[base + i] = (x[base + i] - mean) * rstd * g[i] + b[i];
}

__global__ void k_add_selu(float* __restrict__ a, const float* __restrict__ c, long n){
  long i = (long)blockIdx.x * blockDim.x + threadIdx.x;
  if (i < n) a[i] = selu_f(a[i] + c[i]);
}

__global__ void k_zero(float* __restrict__ p, long n){
  long i = (long)blockIdx.x * blockDim.x + threadIdx.x;
  if (i < n) p[i] = 0.f;
}

// ---------------- capsule kernels ----------------
// vin: [BI][Din*4][H][W] (channel = d*4+r).  capn: [BI][4][Ho][Wo]
__global__ void k_capnorm_pool(const float* __restrict__ x, float* __restrict__ out,
                               int BI, int Din, int H, int W, int K, int stride,
                               int Ho, int Wo){
  long idx = (long)blockIdx.x * blockDim.x + threadIdx.x;
  long total = (long)BI * 4 * Ho * Wo;
  if (idx >= total) return;
  int wo = idx % Wo; long q = idx / Wo; int ho = q % Ho; q /= Ho;
  int r = q & 3; long bi = q >> 2;
  float mx = -3.4e38f;
  for (int ky = 0; ky < K; ++ky){
    int h = ho * stride + ky; if (h >= H) continue;
    for (int kx = 0; kx < K; ++kx){
      int w = wo * stride + kx; if (w >= W) continue;
      float ss = 0.f;
      for (int d = 0; d < Din; ++d){
        float v = x[((bi * (Din * 4) + d * 4 + r) * H + h) * W + w];
        ss += v * v;
      }
      mx = fmaxf(mx, sqrtf(ss));
    }
  }
  out[idx] = mx;
}

// pred layout: [BI][J*16*4][P], channel = (j*16+d)*4+r
__global__ void k_alpha_scale(float* __restrict__ pred, const float* __restrict__ capn,
                              int B, int I, int J, int D, int P){
  long idx = (long)blockIdx.x * blockDim.x + threadIdx.x;
  long total = (long)B * I * J * 4 * P;
  if (idx >= total) return;
  int p = idx % P; long q = idx / P;
  int r = q & 3;  q >>= 2;
  int j = q % J;  q /= J;
  int i = q % I;  int b = q / I;
  long bi = (long)b * I + i;
  long OC = (long)J * D * 4;
  long base = (bi * OC + (long)j * D * 4 + r) * P + p;   // d-stride = 4*P
  float ss = 0.f;
  for (int d = 0; d < D; ++d){ float v = pred[base + (long)d * 4 * P]; ss += v * v; }
  float a = fminf(sqrtf(ss), capn[(bi * 4 + r) * P + p]);
  for (int d = 0; d < D; ++d) pred[base + (long)d * 4 * P] *= a;
}

// softmax over j.  b_ij/c_ij: [BI][J][4*P]
__global__ void k_softmax_j(const float* __restrict__ bij, float* __restrict__ cij,
                            int BI, int J, int P4){
  long idx = (long)blockIdx.x * blockDim.x + threadIdx.x;
  long total = (long)BI * P4;
  if (idx >= total) return;
  long bi = idx / P4; int rp = idx % P4;
  long base = bi * (long)J * P4 + rp;
  float mx = -3.4e38f;
  for (int j = 0; j < J; ++j) mx = fmaxf(mx, bij[base + (long)j * P4]);
  float se = 0.f;
  for (int j = 0; j < J; ++j) se += expf(bij[base + (long)j * P4] - mx);
  float inv = 1.f / se;
  for (int j = 0; j < J; ++j)
    cij[base + (long)j * P4] = expf(bij[base + (long)j * P4] - mx) * inv;
}

// s_j[b,j,d,r,p] = sum_i c_ij * pred   -> sv: [(b*J+j)][(d*4+r)][P]
__global__ void k_sj(const float* __restrict__ cij, const float* __restrict__ pred,
                     float* __restrict__ sv, int B, int I, int J, int D, int P){
  long idx = (long)blockIdx.x * blockDim.x + threadIdx.x;
  long total = (long)B * J * D * 4 * P;
  if (idx >= total) return;
  int p = idx % P; long q = idx / P;
  int r = q & 3;  q >>= 2;
  int d = q % D;  q /= D;
  int j = q % J;  int b = q / J;
  long OC = (long)J * D * 4;
  int P4 = 4 * P;
  float s = 0.f;
  for (int i = 0; i < I; ++i){
    long bi = (long)b * I + i;
    float c  = cij[(bi * J + j) * (long)P4 + r * P + p];
    float pr = pred[(bi * OC + ((long)j * D + d) * 4 + r) * P + p];
    s += c * pr;
  }
  sv[idx] = s;
}

__global__ void k_squash(float* __restrict__ sv, int B, int J, int D, int P){
  long idx = (long)blockIdx.x * blockDim.x + threadIdx.x;
  long total = (long)B * J * 4 * P;
  if (idx >= total) return;
  int p = idx % P; long q = idx / P;
  int r = q & 3;  q >>= 2;
  int j = q % J;  int b = q / J;
  long base = (((long)b * J + j) * (D * 4) + r) * P + p;
  float ss = 0.f;
  for (int d = 0; d < D; ++d){ float v = sv[base + (long)d * 4 * P]; ss += v * v; }
  float nn = sqrtf(ss);
  float f = ss / ((1.f + ss) * (nn + 1e-8f));
  for (int d = 0; d < D; ++d) sv[base + (long)d * 4 * P] *= f;
}

__global__ void k_delta(const float* __restrict__ pred, const float* __restrict__ sv,
                        float* __restrict__ bij, int B, int I, int J, int D, int P){
  long idx = (long)blockIdx.x * blockDim.x + threadIdx.x;
  long total = (long)B * I * J * 4 * P;
  if (idx >= total) return;
  int p = idx % P; long q = idx / P;
  int r = q & 3;  q >>= 2;
  int j = q % J;  q /= J;
  int i = q % I;  int b = q / I;
  long bi = (long)b * I + i;
  long OC = (long)J * D * 4;
  long pbase = (bi * OC + (long)j * D * 4 + r) * P + p;
  long vbase = (((long)b * J + j) * (D * 4) + r) * P + p;
  float ss = 0.f;
  for (int d = 0; d < D; ++d){
    float dv = pred[pbase + (long)d * 4 * P] - sv[vbase + (long)d * 4 * P];
    ss += dv * dv;
  }
  bij[(bi * J + j) * (long)(4 * P) + r * P + p] = sqrtf(ss);
}

__global__ void k_bscale(float* __restrict__ bij, int BI, int J, int P4, float lsc){
  long idx = (long)blockIdx.x * blockDim.x + threadIdx.x;
  long total = (long)BI * P4;
  if (idx >= total) return;
  long bi = idx / P4; int rp = idx % P4;
  long base = bi * (long)J * P4 + rp;
  float m = 0.f;
  for (int j = 0; j < J; ++j) m += bij[base + (long)j * P4];
  m /= (float)J;
  float sc = lsc / (-0.5f * m);
  for (int j = 0; j < J; ++j) bij[base + (long)j * P4] *= sc;
}

// deterministic per-batch entropy: eb[b] = mean_{i,r,p}( -sum_j c ln c / ln J )
__global__ void k_entropy(const float* __restrict__ cij, float* __restrict__ eb,
                          int I, int J, int P4, float invlogJ){
  int b = blockIdx.x;
  long n = (long)I * P4;
  float s = 0.f;
  for (long i = threadIdx.x; i < n; i += blockDim.x){
    long ii = i / P4; long rp = i % P4;
    long base = (((long)b * I + ii) * J) * (long)P4 + rp;
    float e = 0.f;
    for (int j = 0; j < J; ++j){
      float c = cij[base + (long)j * P4];
      e -= c * logf(c);
    }
    s += e * invlogJ;
  }
  __shared__ float sh[256];
  sh[threadIdx.x] = s; __syncthreads();
  for (int o = blockDim.x >> 1; o > 0; o >>= 1){
    if ((int)threadIdx.x < o) sh[threadIdx.x] += sh[threadIdx.x + o];
    __syncthreads();
  }
  if (threadIdx.x == 0) eb[b] = sh[0] / (float)n;
}

// head: sv3 [(b*10+c)*64 + d*4 + r]; out[b*10+c] = max_r (sum_d v*linw[d] + linb)
__global__ void k_head(const float* __restrict__ sv3, const float* __restrict__ linw,
                       const float* __restrict__ linb, const float* __restrict__ eb,
                       float* __restrict__ out, int B){
  int t = blockIdx.x * blockDim.x + threadIdx.x;
  if (t < B * 10){
    long base = (long)t * 64;
    float best = -3.4e38f;
    for (int r = 0; r < 4; ++r){
      float s = linb[0];
      for (int d = 0; d < 16; ++d) s += sv3[base + d * 4 + r] * linw[d];
      best = fmaxf(best, s);
    }
    out[t] = best;
  } else if (t == B * 10){
    float s = 0.f;
    for (int i = 0; i < 96; ++i) s += eb[i];
    out[B * 10] = s;
  }
}

// ============================================================================
// host side
// ============================================================================
static inline int cdiv(int a, int b){ return (a + b - 1) / b; }

static void run_conv(const float* in, const float* wexpT, const float* bias, float* out,
                     int BT, int CIN, int H, int W, int OC, int K, int stride, int pad,
                     int Ho, int Wo, int selu, hipStream_t s){
  dim3 grid(cdiv(BT * Ho * Wo, 16), OC / 64);
  if (pad > 0){
    // only the 3x3 resnet convs are padded
    k_conv_wmma<3, 1><<<grid, 32, 0, s>>>(in, wexpT, bias, out, BT, CIN, H, W, OC,
                                          stride, pad, Ho, Wo, selu);
    return;
  }
  switch (K){
    case 1:
      k_conv_wmma<1, 0><<<grid, 32, 0, s>>>(in, wexpT, bias, out, BT, CIN, H, W, OC,
                                            stride, pad, Ho, Wo, selu);
      break;
    case 3:
      k_conv_wmma<3, 0><<<grid, 32, 0, s>>>(in, wexpT, bias, out, BT, CIN, H, W, OC,
                                            stride, pad, Ho, Wo, selu);
      break;
    default:
      k_conv_wmma<5, 0><<<grid, 32, 0, s>>>(in, wexpT, bias, out, BT, CIN, H, W, OC,
                                            stride, pad, Ho, Wo, selu);
      break;
  }
}

static void basic_block(const float* xin, float* t1, float* t2, float* tsc, float* wexp,
                        int Ci, int Co, int Hi, int stride,
                        const float* c1, const float* g1, const float* b1,
                        const float* c2, const float* g2, const float* b2,
                        const float* wsc, const float* gsc, const float* bsc,
                        hipStream_t s){
  const int CIN = Ci * 4, OC = Co * 4;
  const int Ho = (Hi + 2 - 3) / stride + 1;
  int tot = Co * 4 * Ci * 4 * 9;
  k_expand_p4<<<cdiv(tot, 256), 256, 0, s>>>(c1, wexp, Co, Ci, 3);
  run_conv(xin, wexp, nullptr, t1, 32, CIN, Hi, Hi, OC, 3, stride, 1, Ho, Ho, 0, s);
  k_bn_fused<<<Co, 256, 0, s>>>(t1, g1, b1, 32, Co, Ho * Ho, 1);
  tot = Co * 4 * Co * 4 * 9;
  k_expand_p4<<<cdiv(tot, 256), 256, 0, s>>>(c2, wexp, Co, Co, 3);
  run_conv(t1, wexp, nullptr, t2, 32, OC, Ho, Ho, OC, 3, 1, 1, Ho, Ho, 0, s);
  k_bn_fused<<<Co, 256, 0, s>>>(t2, g2, b2, 32, Co, Ho * Ho, 0);
  const float* scp = xin;
  if (wsc){
    tot = Co * 4 * Ci * 4;
    k_expand_p4<<<cdiv(tot, 256), 256, 0, s>>>(wsc, wexp, Co, Ci, 1);
    run_conv(xin, wexp, nullptr, tsc, 32, CIN, Hi, Hi, OC, 1, stride, 0, Ho, Ho, 0, s);
    k_bn_fused<<<Co, 256, 0, s>>>(tsc, gsc, bsc, 32, Co, Ho * Ho, 0);
    scp = tsc;
  }
  long n = (long)32 * OC * Ho * Ho;
  k_add_selu<<<cdiv((int)n, 256), 256, 0, s>>>(t2, scp, n);   // result in t2
}

static void caps_layer(const float* vin, int B, int I, int Din, int J, int K, int stride,
                       int H, int W,
                       const float* w, const float* bias, const float* g, const float* beta,
                       float* wexp, float* pred, float* bb, float* cb, float* sv,
                       float* capn, float* eb_slot, hipStream_t s){
  const int D = 16;
  const int Ho = (H - K) / stride + 1, Wo = (W - K) / stride + 1, P = Ho * Wo;
  const int Co = J * D;
  int tot = Co * 4 * Din * 4 * K * K;
  k_expand_p4<<<cdiv(tot, 256), 256, 0, s>>>(w, wexp, Co, Din, K);
  const int BT = B * I, CIN = Din * 4, OC = Co * 4;
  run_conv(vin, wexp, bias, pred, BT, CIN, H, W, OC, K, stride, 0, Ho, Wo, 0, s);
  k_bn_fused<<<Co, 256, 0, s>>>(pred, g, beta, BT, Co, P, 0);
  {
    long tt = (long)BT * 4 * P;
    k_capnorm_pool<<<cdiv((int)tt, 256), 256, 0, s>>>(vin, capn, BT, Din, H, W, K,
                                                      stride, Ho, Wo);
  }
  {
    long tt = (long)B * I * J * 4 * P;
    k_alpha_scale<<<cdiv((int)tt, 256), 256, 0, s>>>(pred, capn, B, I, J, D, P);
  }
  const long nb = (long)BT * J * 4 * P;
  k_zero<<<cdiv((int)nb, 256), 256, 0, s>>>(bb, nb);
  const int P4 = 4 * P;
  const float lsc = logf(0.9f * (float)J) - logf(0.1f);
  for (int it = 0; it < 3; ++it){
    k_softmax_j<<<cdiv(BT * P4, 256), 256, 0, s>>>(bb, cb, BT, J, P4);
    long ns = (long)B * J * D * 4 * P;
    k_sj<<<cdiv((int)ns, 256), 256, 0, s>>>(cb, pred, sv, B, I, J, D, P);
    long nq = (long)B * J * 4 * P;
    k_squash<<<cdiv((int)nq, 256), 256, 0, s>>>(sv, B, J, D, P);
    if (it < 2){
      long nd = (long)BT * J * 4 * P;
      k_delta<<<cdiv((int)nd, 256), 256, 0, s>>>(pred, sv, bb, B, I, J, D, P);
      k_bscale<<<cdiv(BT * P4, 256), 256, 0, s>>>(bb, BT, J, P4, lsc);
    }
  }
  k_entropy<<<B, 256, 0, s>>>(cb, eb_slot, I, J, P4, 1.f / logf((float)J));
}

extern "C" void kernel_launch(void* const* d_in, const int* in_sizes, int n_in,
                              void* d_out, int out_size, void* d_ws, size_t ws_size,
                              hipStream_t stream){
  (void)in_sizes; (void)n_in; (void)out_size; (void)ws_size;
  const float** in = (const float**)d_in;
  float* out = (float*)d_out;

  // pytree flatten order (sorted dict keys, tuples in declaration order):
  const float *bn1_g = in[0], *bn1_b = in[1];
  const float *caps1_bias = in[2], *caps1_g = in[3], *caps1_beta = in[4], *caps1_w = in[5];
  const float *caps2_bias = in[6], *caps2_g = in[7], *caps2_beta = in[8], *caps2_w = in[9];
  const float *caps3_bias = in[10], *caps3_g = in[11], *caps3_beta = in[12], *caps3_w = in[13];
  const float *conv1_w = in[14];
  const float *l1b1_g1 = in[15], *l1b1_b1 = in[16], *l1b1_g2 = in[17], *l1b1_b2 = in[18],
              *l1b1_c1 = in[19], *l1b1_c2 = in[20];
  const float *l1b2_g1 = in[21], *l1b2_b1 = in[22], *l1b2_g2 = in[23], *l1b2_b2 = in[24],
              *l1b2_c1 = in[25], *l1b2_c2 = in[26];
  const float *l2b1_g1 = in[27], *l2b1_b1 = in[28], *l2b1_g2 = in[29], *l2b1_b2 = in[30],
              *l2b1_c1 = in[31], *l2b1_c2 = in[32], *l2b1_sc = in[33],
              *l2b1_gs = in[34], *l2b1_bs = in[35];
  const float *l2b2_g1 = in[36], *l2b2_b1 = in[37], *l2b2_g2 = in[38], *l2b2_b2 = in[39],
              *l2b2_c1 = in[40], *l2b2_c2 = in[41];
  const float *lin_b = in[42], *lin_w = in[43];
  const float *prim_b = in[44], *prim_lnb = in[45], *prim_lng = in[46], *prim_w = in[47];
  const float *x = in[48];

  // workspace carve (floats)
  float* base = (float*)d_ws;
  size_t off = 0;
  float* wexp = base + off; off += 1024000;     // max expanded weight (caps3)
  float* actA = base + off; off += 2097152;
  float* actB = base + off; off += 2097152;
  float* actC = base + off; off += 2097152;
  float* actS = base + off; off += 2097152;
  float* pc   = base + off; off += 4194304;     // prim out  (32,512,16,16)
  float* pred = base + off; off += 51380224;    // caps1 pred (largest)
  float* bb   = base + off; off += 3211264;
  float* cb   = base + off; off += 3211264;
  float* sv   = base + off; off += 3211264;
  float* capn = base + off; off += 131072;
  float* eb   = base + off; off += 96;

  // ---- conv1 (z2) + bn + selu -> actA : [32][64][32][32]
  k_expand_z2<<<cdiv(16 * 4 * 9, 256), 256, 0, stream>>>(conv1_w, wexp, 16, 3);
  run_conv(x, wexp, nullptr, actA, 32, 1, 32, 32, 64, 3, 1, 1, 32, 32, 0, stream);
  k_bn_fused<<<16, 256, 0, stream>>>(actA, bn1_g, bn1_b, 32, 16, 1024, 1);

  // ---- resnet blocks
  basic_block(actA, actB, actC, actS, wexp, 16, 16, 32, 1,
              l1b1_c1, l1b1_g1, l1b1_b1, l1b1_c2, l1b1_g2, l1b1_b2,
              nullptr, nullptr, nullptr, stream);                 // out: actC (32x64x32x32)
  basic_block(actC, actA, actB, actS, wexp, 16, 16, 32, 1,
              l1b2_c1, l1b2_g1, l1b2_b1, l1b2_c2, l1b2_g2, l1b2_b2,
              nullptr, nullptr, nullptr, stream);                 // out: actB
  basic_block(actB, actA, actC, actS, wexp, 16, 32, 32, 2,
              l2b1_c1, l2b1_g1, l2b1_b1, l2b1_c2, l2b1_g2, l2b1_b2,
              l2b1_sc, l2b1_gs, l2b1_bs, stream);                 // out: actC (32x128x16x16)
  basic_block(actC, actA, actB, actS, wexp, 32, 32, 16, 1,
              l2b2_c1, l2b2_g1, l2b2_b1, l2b2_c2, l2b2_g2, l2b2_b2,
              nullptr, nullptr, nullptr, stream);                 // out: actB

  // ---- prim caps: 1x1 p4conv + bias + selu, then per-batch layernorm
  k_expand_p4<<<cdiv(512 * 128, 256), 256, 0, stream>>>(prim_w, wexp, 128, 32, 1);
  run_conv(actB, wexp, prim_b, pc, 32, 128, 16, 16, 512, 1, 1, 0, 16, 16, 1, stream);
  k_layernorm<<<32, 256, 0, stream>>>(pc, prim_lng, prim_lnb, 131072);

  // ---- capsule layers (pc layout == caps1 conv-input layout)
  caps_layer(pc, 32, 16, 8, 32, 3, 2, 16, 16,
             caps1_w, caps1_bias, caps1_g, caps1_beta,
             wexp, pred, bb, cb, sv, capn, eb + 0, stream);       // v -> sv (B,32,16,4,7,7)
  caps_layer(sv, 32, 32, 16, 16, 3, 1, 7, 7,
             caps2_w, caps2_bias, caps2_g, caps2_beta,
             wexp, pred, bb, cb, sv, capn, eb + 32, stream);      // v -> sv (B,16,16,4,5,5)
  caps_layer(sv, 32, 16, 16, 10, 5, 2, 5, 5,
             caps3_w, caps3_bias, caps3_g, caps3_beta,
             wexp, pred, bb, cb, sv, capn, eb + 64, stream);      // v -> sv (B,10,16,4,1,1)

  // ---- head: out[0..319] = logits max over r, out[320] = e1+e2+e3
  k_head<<<1, 352, 0, stream>>>(sv, lin_w, lin_b, eb, out, 32);
}